// GATForPortT5_81819126989064
// MI455X (gfx1250) — compile-verified
//
#include <hip/hip_runtime.h>
#include <hip/hip_bf16.h>
#include <math.h>

#define NNODES 50000
#define NEDGES 800000
#define DDIM   256

typedef float v2f __attribute__((ext_vector_type(2)));
typedef float v8f __attribute__((ext_vector_type(8)));

// ======================= WMMA GEMM: Y = X@W (+bias) (+addend) ================
// X:[M,256] row-major, W:[256,256] row-major (W[k*256+n]), Y:[M,256].
// Block = 128 threads (4 waves). Each block computes a 64(M) x 32(N) tile:
// wave w -> rows [mtile*16, +16), two 16x16 N-tiles with separate accumulators
// (A fragment reused by 2 WMMAs). W column tile (256x32) is staged once per
// block into LDS *transposed* (WT[n][k], row stride 258 to avoid bank
// conflicts), so each B fragment is a single ds_load_b64 of K-adjacent pair.
// K stepped by 4 via V_WMMA_F32_16X16X4_F32 (fp32 in/accum: precision-safe).
// A-frag layout (ISA 7.12.2, 32-bit A 16x4): lanes0-15=M, VGPR0 K=0/K=2,
// VGPR1 K=1/K=3.  B/C/D: row striped across lanes within a VGPR.
#define WT_STRIDE 258
__global__ void __launch_bounds__(128)
gemm256_wmma(const float* __restrict__ X, const float* __restrict__ W,
             const float* __restrict__ bias, const float* __restrict__ addend,
             float* __restrict__ Y, int Mtiles) {
  __shared__ float wt[32 * WT_STRIDE];             // 33 KB
  const int tid   = threadIdx.x;
  const int wave  = tid >> 5;                      // wave32
  const int lane  = tid & 31;
  const int nbase = blockIdx.y * 32;

  // ---- stage W[:, nbase:nbase+32] transposed into LDS (all 128 threads) ----
  // iter i: k = i*4 + wave, n = lane  -> coalesced 128B global read per wave
#pragma unroll 4
  for (int i = 0; i < 64; ++i) {
    const int k = i * 4 + wave;
    wt[lane * WT_STRIDE + k] = W[(size_t)k * DDIM + nbase + lane];
  }
  __syncthreads();

  const int mtile = blockIdx.x * 4 + wave;
  if (mtile >= Mtiles) return;                     // uniform per wave -> EXEC all-1

  const int l16  = lane & 15;
  const int half = lane >> 4;
  const int mrow = mtile * 16;
  const float* Arow = X + (size_t)(mrow + l16) * DDIM;
  const float* wt0  = wt + (size_t)l16 * WT_STRIDE;        // N-tile 0 column
  const float* wt1  = wt + (size_t)(16 + l16) * WT_STRIDE; // N-tile 1 column

  v8f acc0 = {0.f, 0.f, 0.f, 0.f, 0.f, 0.f, 0.f, 0.f};
  v8f acc1 = {0.f, 0.f, 0.f, 0.f, 0.f, 0.f, 0.f, 0.f};
#pragma unroll 8
  for (int kb = 0; kb < DDIM; kb += 4) {
    const int k0 = kb + 2 * half;
    const v2f a  = *(const v2f*)(Arow + k0);       // global b64, contiguous K pair
    const v2f b0 = *(const v2f*)(wt0 + k0);        // ds_load_b64
    const v2f b1 = *(const v2f*)(wt1 + k0);        // ds_load_b64
    acc0 = __builtin_amdgcn_wmma_f32_16x16x4_f32(false, a, false, b0,
                                                 (short)0, acc0, false, false);
    acc1 = __builtin_amdgcn_wmma_f32_16x16x4_f32(false, a, false, b1,
                                                 (short)0, acc1, false, false);
  }

  // ---- epilogue: bias + optional residual add, branch hoisted out ----------
  const int nc0 = nbase + l16;
  const int nc1 = nc0 + 16;
  const float bb0 = bias ? bias[nc0] : 0.f;
  const float bb1 = bias ? bias[nc1] : 0.f;
  if (addend) {
#pragma unroll
    for (int r = 0; r < 8; ++r) {
      const size_t ro = (size_t)(mrow + r + 8 * half) * DDIM;
      Y[ro + nc0] = acc0[r] + bb0 + addend[ro + nc0];
      Y[ro + nc1] = acc1[r] + bb1 + addend[ro + nc1];
    }
  } else {
#pragma unroll
    for (int r = 0; r < 8; ++r) {
      const size_t ro = (size_t)(mrow + r + 8 * half) * DDIM;
      Y[ro + nc0] = acc0[r] + bb0;
      Y[ro + nc1] = acc1[r] + bb1;
    }
  }
}

// ================= per-node attention dots: ai[n,h], aj[n,h] =================
__global__ void node_dots(const float* __restrict__ h, const float* __restrict__ att,
                          float* __restrict__ ai, float* __restrict__ aj,
                          int N, int H) {
  const int t = blockIdx.x * blockDim.x + threadIdx.x;
  if (t >= N * H) return;
  const int n = t / H, hh = t % H;
  const int C = DDIM / H;
  const float* hp = h + (size_t)n * DDIM + hh * C;
  const float* wi = att + (size_t)hh * 3 * C;      // att_i
  const float* wj = wi + C;                        // att_j
  float si = 0.f, sj = 0.f;
  for (int c = 0; c < C; ++c) { const float v = hp[c]; si += v * wi[c]; sj += v * wj[c]; }
  ai[t] = si; aj[t] = sj;
}

// ================= per-head edge scale: s_e[h] = dot(edgeW_h, att_e_h) =======
__global__ void edge_scale(const float* __restrict__ edgeW, const float* __restrict__ att,
                           float* __restrict__ s_e, int H) {
  const int hh = threadIdx.x;
  if (hh >= H) return;
  const int C = DDIM / H;
  const float* we = att + (size_t)hh * 3 * C + 2 * C;  // att_e
  float s = 0.f;
  for (int c = 0; c < C; ++c) s += edgeW[hh * C + c] * we[c];
  s_e[hh] = s;
}

// ======================= segment softmax helpers =============================
__device__ __forceinline__ void atomicMaxFloat(float* addr, float v) {
  // monotone bit trick: int compare matches float for >=0; reversed uint for <0
  if (v >= 0.f) atomicMax((int*)addr, __float_as_int(v));
  else          atomicMin((unsigned int*)addr, (unsigned int)__float_as_int(v));
}

__device__ __forceinline__ float edge_alpha(const float* ai, const float* aj,
                                            const float* ea, const float* s_e,
                                            int r, int c, int e, int hh, int H) {
  float a = ai[r * H + hh] + aj[c * H + hh] + ea[e] * s_e[hh];
  return a > 0.f ? a : 0.2f * a;                   // leaky_relu(0.2)
}

__global__ void init_nh(float* __restrict__ amax, float* __restrict__ denom, int NH) {
  const int t = blockIdx.x * blockDim.x + threadIdx.x;
  if (t >= NH) return;
  amax[t]  = -INFINITY;
  denom[t] = 0.f;
}

__global__ void edge_max(const float* __restrict__ ai, const float* __restrict__ aj,
                         const float* __restrict__ ea, const float* __restrict__ s_e,
                         const int* __restrict__ row, const int* __restrict__ col,
                         float* __restrict__ amax, int E, int H) {
  const int t = blockIdx.x * blockDim.x + threadIdx.x;
  if (t >= E * H) return;
  const int e = t / H, hh = t % H;
  const int r = row[e], c = col[e];
  atomicMaxFloat(&amax[r * H + hh], edge_alpha(ai, aj, ea, s_e, r, c, e, hh, H));
}

__global__ void fix_amax(float* __restrict__ amax, int NH) {
  const int t = blockIdx.x * blockDim.x + threadIdx.x;
  if (t >= NH) return;
  const float v = amax[t];
  if (!isfinite(v)) amax[t] = 0.f;                 // nodes with no in-edges
}

__global__ void edge_expsum(const float* __restrict__ ai, const float* __restrict__ aj,
                            const float* __restrict__ ea, const float* __restrict__ s_e,
                            const int* __restrict__ row, const int* __restrict__ col,
                            const float* __restrict__ amax, float* __restrict__ ex,
                            float* __restrict__ denom, int E, int H) {
  const int t = blockIdx.x * blockDim.x + threadIdx.x;
  if (t >= E * H) return;
  const int e = t / H, hh = t % H;
  const int r = row[e], c = col[e];
  const float a  = edge_alpha(ai, aj, ea, s_e, r, c, e, hh, H);
  const float xv = expf(a - amax[r * H + hh]);
  ex[t] = xv;
  atomicAdd(&denom[r * H + hh], xv);
}

__global__ void zero_f(float* __restrict__ p, int n) {
  const int t = blockIdx.x * blockDim.x + threadIdx.x;
  if (t < n) p[t] = 0.f;
}

// out[row] += softmax(alpha) * h[col]  (one thread = one edge x 4 features)
__global__ void edge_scatter(const float* __restrict__ hmat, const float* __restrict__ ex,
                             const float* __restrict__ denom,
                             const int* __restrict__ row, const int* __restrict__ col,
                             float* __restrict__ out, int E, int H) {
  const long long t = (long long)blockIdx.x * blockDim.x + threadIdx.x;
  if (t >= (long long)E * 64) return;
  const int d4 = (int)(t & 63);                    // feature group (4 floats)
  const int e  = (int)(t >> 6);
  const int C  = DDIM / H;
  const int hh = (d4 * 4) / C;
  const int r = row[e], c = col[e];
  const float a = ex[(size_t)e * H + hh] / (denom[r * H + hh] + 1e-16f);
  const float4 hv = *(const float4*)(hmat + (size_t)c * DDIM + d4 * 4);
  float* o = out + (size_t)r * DDIM + d4 * 4;
  atomicAdd(o + 0, a * hv.x);
  atomicAdd(o + 1, a * hv.y);
  atomicAdd(o + 2, a * hv.z);
  atomicAdd(o + 3, a * hv.w);
}

// ===================== normalization epilogues ===============================
__global__ void bn_relu(float* __restrict__ x, const float* __restrict__ g,
                        const float* __restrict__ b, int n) {
  const int t = blockIdx.x * blockDim.x + threadIdx.x;
  if (t >= n) return;
  const int cc = t & (DDIM - 1);
  const float scale = g[cc] * rsqrtf(1.f + 1e-5f); // running_mean=0, var=1
  const float v = x[t] * scale + b[cc];
  x[t] = v > 0.f ? v : 0.f;
}

// one wave32 per node: LN over 256 features + ReLU, in place
__global__ void ln_relu(float* __restrict__ x, const float* __restrict__ g,
                        const float* __restrict__ b, int N) {
  const int wid  = (blockIdx.x * blockDim.x + threadIdx.x) >> 5;
  const int lane = threadIdx.x & 31;
  if (wid >= N) return;
  float* xp = x + (size_t)wid * DDIM;
  float vals[8];
  float s = 0.f;
#pragma unroll
  for (int i = 0; i < 8; ++i) { vals[i] = xp[lane + 32 * i]; s += vals[i]; }
#pragma unroll
  for (int o = 16; o > 0; o >>= 1) s += __shfl_xor(s, o, 32);
  const float mean = s * (1.f / DDIM);
  float vs = 0.f;
#pragma unroll
  for (int i = 0; i < 8; ++i) { const float d = vals[i] - mean; vs += d * d; }
#pragma unroll
  for (int o = 16; o > 0; o >>= 1) vs += __shfl_xor(vs, o, 32);
  const float inv = rsqrtf(vs * (1.f / DDIM) + 1e-5f);
#pragma unroll
  for (int i = 0; i < 8; ++i) {
    const int cc = lane + 32 * i;
    const float y = (vals[i] - mean) * inv * g[cc] + b[cc];
    xp[cc] = y > 0.f ? y : 0.f;
  }
}

// =============================== host side ===================================
static void run_block(const float* xin, float* xout,
                      const float* linW, const float* linb,
                      const float* gatlin, const float* gatedge, const float* gatatt,
                      const float* norm_g, const float* norm_b,
                      const float* skipW, const float* skipb,
                      bool useLN, int H,
                      float* B1, float* B2, float* ai, float* aj,
                      float* amax, float* denom, float* ex, float* s_e,
                      const int* row, const int* col, const float* eattr,
                      hipStream_t stream) {
  const int N = NNODES, E = NEDGES;
  const int Mtiles = N / 16;                       // 50000 % 16 == 0
  const dim3 gg((Mtiles + 3) / 4, DDIM / 32);      // 4 waves x 16 rows, 32 cols
  const int NH = N * H, EH = E * H;

  // h_lin = xin@linW + linb ; h = h_lin@gatlin
  gemm256_wmma<<<gg, 128, 0, stream>>>(xin, linW, linb, nullptr, B1, Mtiles);
  gemm256_wmma<<<gg, 128, 0, stream>>>(B1, gatlin, nullptr, nullptr, B2, Mtiles);
  // attention scalars
  node_dots<<<(NH + 255) / 256, 256, 0, stream>>>(B2, gatatt, ai, aj, N, H);
  edge_scale<<<1, 32, 0, stream>>>(gatedge, gatatt, s_e, H);
  // segment softmax
  init_nh<<<(NH + 255) / 256, 256, 0, stream>>>(amax, denom, NH);
  edge_max<<<(EH + 255) / 256, 256, 0, stream>>>(ai, aj, eattr, s_e, row, col, amax, E, H);
  fix_amax<<<(NH + 255) / 256, 256, 0, stream>>>(amax, NH);
  edge_expsum<<<(EH + 255) / 256, 256, 0, stream>>>(ai, aj, eattr, s_e, row, col,
                                                    amax, ex, denom, E, H);
  // aggregate into B1 (h_lin no longer needed -> reuse as accumulator)
  zero_f<<<(N * DDIM + 255) / 256, 256, 0, stream>>>(B1, N * DDIM);
  edge_scatter<<<(int)(((long long)E * 64 + 255) / 256), 256, 0, stream>>>(
      B2, ex, denom, row, col, B1, E, H);
  // norm + relu
  if (useLN) ln_relu<<<(N * 32 + 255) / 256, 256, 0, stream>>>(B1, norm_g, norm_b, N);
  else       bn_relu<<<(N * DDIM + 255) / 256, 256, 0, stream>>>(B1, norm_g, norm_b, N * DDIM);
  // xout = B1 + xin@skipW + skipb  (fused in GEMM epilogue)
  gemm256_wmma<<<gg, 128, 0, stream>>>(xin, skipW, skipb, B1, xout, Mtiles);
}

extern "C" void kernel_launch(void* const* d_in, const int* in_sizes, int n_in,
                              void* d_out, int out_size, void* d_ws, size_t ws_size,
                              hipStream_t stream) {
  (void)in_sizes; (void)n_in; (void)out_size; (void)ws_size;
  const float* x     = (const float*)d_in[0];
  const int*   eidx  = (const int*)d_in[1];
  const float* eattr = (const float*)d_in[2];
  const int* row = eidx;                 // edge_index[0]
  const int* col = eidx + NEDGES;        // edge_index[1]

  // params in dict insertion order starting at d_in[3]
  auto P = [&](int i) { return (const float*)d_in[3 + i]; };
  // block1: 0 lin1_W 1 lin1_b 2 skip1_W 3 skip1_b 4 gat1_lin 5 gat1_edge 6 gat1_att 7 bn1_g 8 bn1_b
  // block2: 9..17   block3: 18 lin3_W 19 lin3_b 20 skip3_W 21 skip3_b 22 gat3_lin
  //                          23 gat3_edge 24 gat3_att 25 ln_g 26 ln_b

  float* ws = (float*)d_ws;
  float* X1    = ws; ws += (size_t)NNODES * DDIM;
  float* X2    = ws; ws += (size_t)NNODES * DDIM;
  float* B1    = ws; ws += (size_t)NNODES * DDIM;
  float* B2    = ws; ws += (size_t)NNODES * DDIM;
  float* ai    = ws; ws += (size_t)NNODES * 8;
  float* aj    = ws; ws += (size_t)NNODES * 8;
  float* amax  = ws; ws += (size_t)NNODES * 8;
  float* denom = ws; ws += (size_t)NNODES * 8;
  float* ex    = ws; ws += (size_t)NEDGES * 8;
  float* s_e   = ws; ws += 8;

  // block 1 (H=8, BN)
  run_block(x,  X1, P(0),  P(1),  P(4),  P(5),  P(6),  P(7),  P(8),  P(2),  P(3),
            false, 8, B1, B2, ai, aj, amax, denom, ex, s_e, row, col, eattr, stream);
  // block 2 (H=8, BN)
  run_block(X1, X2, P(9),  P(10), P(13), P(14), P(15), P(16), P(17), P(11), P(12),
            false, 8, B1, B2, ai, aj, amax, denom, ex, s_e, row, col, eattr, stream);
  // block 3 (H=1, LN) -> d_out
  run_block(X2, (float*)d_out, P(18), P(19), P(22), P(23), P(24), P(25), P(26),
            P(20), P(21), true, 1, B1, B2, ai, aj, amax, denom, ex, s_e,
            row, col, eattr, stream);
}